// MultiHeadGraphAttention_15857019257033
// MI455X (gfx1250) — compile-verified
//
#include <hip/hip_runtime.h>

#define N_NODES 50000
#define N_EDGES 800000
#define F_OUT   128
#define N_HEAD  4
#define NEG_SLOPE 0.2f

typedef __attribute__((ext_vector_type(2))) float v2f;
typedef __attribute__((ext_vector_type(8))) float v8f;

__device__ __forceinline__ void atomF(float* p, float v) {
    // Guaranteed hardware global_atomic_add_f32 (no CAS loop) on AMD.
    unsafeAtomicAdd(p, v);
}

// ---------------- zero init ----------------
__global__ void zero_f4(float4* __restrict__ p, int n4) {
    int i = blockIdx.x * blockDim.x + threadIdx.x;
    int stride = gridDim.x * blockDim.x;
    float4 z = make_float4(0.f, 0.f, 0.f, 0.f);
    for (; i < n4; i += stride) p[i] = z;
}

// ---------------- coefficient B matrix: [128 k][16 cols] ----------------
// col 0..3 = w[head]*a_src[head], col 4..7 = w[head]*a_dst[head], col 8..15 = 0
__global__ void coef_kernel(const float* __restrict__ w, const float* __restrict__ a,
                            float* __restrict__ coefB) {
    int idx = blockIdx.x * blockDim.x + threadIdx.x;
    if (idx >= F_OUT * 16) return;
    int k = idx >> 4, col = idx & 15;
    float v = 0.f;
    if (col < 8) {
        int head = col & 3, sel = col >> 2;
        v = w[head * F_OUT + k] * a[head * 2 * F_OUT + sel * F_OUT + k];
    }
    coefB[idx] = v;
}

// ---------------- WMMA projection: s = x @ coefB  (16-node tile per wave) ----------------
__global__ void proj_kernel(const float* __restrict__ x, const float* __restrict__ coefB,
                            float* __restrict__ s_src, float* __restrict__ s_dst) {
    int tile = blockIdx.x;          // 3125 tiles of 16 nodes
    int lane = threadIdx.x;         // 0..31 (wave32)
    int half = lane >> 4;           // 0 or 1
    int rc   = lane & 15;           // A: row M, B: col N
    int koff = half * 2;            // A lanes 16-31 hold K=2,3; B mirrored
    long arow = (long)(tile * 16 + rc) * F_OUT;

    v8f c = {0.f, 0.f, 0.f, 0.f, 0.f, 0.f, 0.f, 0.f};
#pragma unroll
    for (int k = 0; k < F_OUT; k += 4) {
        v2f av, bv;
        float2 ax = *(const float2*)(x + arow + k + koff);
        av.x = ax.x; av.y = ax.y;
        bv.x = coefB[(k + koff) * 16 + rc];
        bv.y = coefB[(k + koff + 1) * 16 + rc];
        c = __builtin_amdgcn_wmma_f32_16x16x4_f32(false, av, false, bv,
                                                  (short)0, c, false, false);
    }
    // C/D layout: VGPR j, lanes 0-15 -> M=j, N=lane; lanes 16-31 -> M=j+8, N=lane-16
    if (rc < 8) {
        int head = rc & 3;
        float* buf = (rc >> 2) ? s_dst : s_src;
        int mbase = tile * 16 + half * 8;
#pragma unroll
        for (int j = 0; j < 8; ++j)
            buf[head * N_NODES + mbase + j] = c[j];
    }
}

// ---------------- edge scatter: one wave per edge, heads looped ----------------
__global__ void edge_kernel(const float* __restrict__ x, const float* __restrict__ w,
                            const int* __restrict__ ei,
                            const float* __restrict__ s_src, const float* __restrict__ s_dst,
                            float* __restrict__ rowsum, float* __restrict__ out) {
    int tid    = blockIdx.x * blockDim.x + threadIdx.x;
    int wave   = tid >> 5;
    int lane   = tid & 31;
    int nwaves = (gridDim.x * blockDim.x) >> 5;
    int f      = lane * 4;

    // Per-lane head weights are loop-invariant: hoist (4 heads x float4).
    float4 wv[N_HEAD];
#pragma unroll
    for (int i = 0; i < N_HEAD; ++i)
        wv[i] = *(const float4*)(w + i * F_OUT + f);

    for (int e = wave; e < N_EDGES; e += nwaves) {
        int sn = ei[e];
        int dn = ei[N_EDGES + e];
        float4 xv = *(const float4*)(x + (long)dn * F_OUT + f);   // reused by all heads
#pragma unroll
        for (int i = 0; i < N_HEAD; ++i) {
            float logit = s_src[i * N_NODES + sn] + s_dst[i * N_NODES + dn];
            float lr  = (logit >= 0.f) ? logit : NEG_SLOPE * logit;
            float val = expf(-lr);
            if (lane == 0) atomF(&rowsum[i * N_NODES + sn], val);
            float* o = out + ((long)i * N_NODES + sn) * F_OUT + f;
            atomF(o + 0, val * xv.x * wv[i].x);
            atomF(o + 1, val * xv.y * wv[i].y);
            atomF(o + 2, val * xv.z * wv[i].z);
            atomF(o + 3, val * xv.w * wv[i].w);
        }
    }
}

// ---------------- normalize ----------------
__global__ void norm_kernel(float4* __restrict__ out, const float* __restrict__ rowsum, int n4) {
    int i = blockIdx.x * blockDim.x + threadIdx.x;
    int stride = gridDim.x * blockDim.x;
    for (; i < n4; i += stride) {
        float r = rowsum[i >> 5];   // 32 float4 = 128 floats per (head,node)
        float4 v = out[i];
        float inv = 1.0f / r;
        v.x *= inv; v.y *= inv; v.z *= inv; v.w *= inv;
        out[i] = v;
    }
}

extern "C" void kernel_launch(void* const* d_in, const int* in_sizes, int n_in,
                              void* d_out, int out_size, void* d_ws, size_t ws_size,
                              hipStream_t stream) {
    const float* x  = (const float*)d_in[0];   // [N, 128]
    const float* w  = (const float*)d_in[1];   // [4, 1, 128]
    const float* a  = (const float*)d_in[2];   // [4, 256, 1]
    const int*   ei = (const int*)d_in[3];     // [2, E]
    float* out = (float*)d_out;                // [4, N, 128]

    float* ws     = (float*)d_ws;
    float* s_src  = ws;                        // 4*N
    float* s_dst  = ws + N_HEAD * N_NODES;     // 4*N
    float* rowsum = ws + 2 * N_HEAD * N_NODES; // 4*N
    float* coefB  = ws + 3 * N_HEAD * N_NODES; // 128*16

    const int out_f4 = N_HEAD * N_NODES * F_OUT / 4;   // 6.4M
    const int rs_f4  = N_HEAD * N_NODES / 4;           // 50k

    zero_f4<<<4096, 256, 0, stream>>>((float4*)out, out_f4);
    zero_f4<<<196, 256, 0, stream>>>((float4*)rowsum, rs_f4);
    coef_kernel<<<8, 256, 0, stream>>>(w, a, coefB);
    proj_kernel<<<N_NODES / 16, 32, 0, stream>>>(x, coefB, s_src, s_dst);
    edge_kernel<<<2048, 256, 0, stream>>>(x, w, ei, s_src, s_dst, rowsum, out);
    norm_kernel<<<4096, 256, 0, stream>>>((float4*)out, rowsum, out_f4);
}